// SuppSelfAttention_7782480740896
// MI455X (gfx1250) — compile-verified
//
#include <hip/hip_runtime.h>
#include <hip/hip_bf16.h>

namespace {

constexpr int B_ = 4, H_ = 64, W_ = 64, N_ = H_ * W_, C_ = 384;
constexpr int HEADS_ = 12, PTS_ = 9, HD_ = 32;
constexpr int KFG_ = 20, KBG_ = 50, NITER_ = 10, LAYERS_ = 2;
constexpr int BN_ = B_ * N_;
constexpr int FFN_ = 3 * C_;              // 1152
constexpr int NOFF_ = HEADS_ * PTS_ * 2;  // 216
constexpr int NAW_ = HEADS_ * PTS_;       // 108
constexpr int NOUT_ROWS_ = N_ + KFG_ + KBG_;  // 4166
constexpr int KB_ = 96;                   // K-slab staged in LDS (divides 384 and 1152)

typedef __attribute__((ext_vector_type(2))) float v2f;
typedef __attribute__((ext_vector_type(8))) float v8f;
typedef __attribute__((ext_vector_type(4))) unsigned int u32x4;
typedef __attribute__((ext_vector_type(8))) int i32x8;
typedef __attribute__((ext_vector_type(4))) int i32x4;

} // namespace

// ---------------------------------------------------------------------------
// Prep: x[b,n,c] = s_x[b,c,n]
// ---------------------------------------------------------------------------
__global__ void transpose_kernel(const float* __restrict__ sx, float* __restrict__ X) {
    size_t i = (size_t)blockIdx.x * blockDim.x + threadIdx.x;
    if (i >= (size_t)BN_ * C_) return;
    int c = (int)(i % C_);
    size_t bn = i / C_;
    int n = (int)(bn % N_);
    int b = (int)(bn / N_);
    X[i] = sx[((size_t)b * C_ + c) * N_ + n];
}

// notpad = 1 - (pad==255); wfg = obj*notpad; wbg = (1-obj)*notpad
__global__ void mask_kernel(const float* __restrict__ padm, const float* __restrict__ suppm,
                            float* __restrict__ notpad, float* __restrict__ wfg,
                            float* __restrict__ wbg) {
    int i = blockIdx.x * blockDim.x + threadIdx.x;
    if (i >= BN_) return;
    float pd = (padm[i] == 255.0f) ? 1.f : 0.f;
    float ob = (suppm[i] == 1.0f) ? 1.f : 0.f;
    float np = 1.f - pd;
    notpad[i] = np;
    wfg[i] = ob * np;
    wbg[i] = (1.f - ob) * np;
}

// ---------------------------------------------------------------------------
// Sine positional encoding (normalize=True), pos[b,n,0:192)=y-part, [192:384)=x
// ---------------------------------------------------------------------------
__global__ void pos_kernel(const float* __restrict__ notpad, float* __restrict__ pos) {
    int i = blockIdx.x * blockDim.x + threadIdx.x;
    if (i >= BN_) return;
    int b = i / N_, n = i % N_;
    int hh = n / W_, ww = n % W_;
    const float* np = notpad + (size_t)b * N_;
    float ycum = 0.f, ylast, xcum = 0.f, xlast;
    for (int r = 0; r <= hh; ++r) ycum += np[r * W_ + ww];
    ylast = ycum;
    for (int r = hh + 1; r < H_; ++r) ylast += np[r * W_ + ww];
    for (int ccol = 0; ccol <= ww; ++ccol) xcum += np[hh * W_ + ccol];
    xlast = xcum;
    for (int ccol = ww + 1; ccol < W_; ++ccol) xlast += np[hh * W_ + ccol];
    const float TWO_PI = 6.283185307179586f;
    float yv = ycum / (ylast + 1e-6f) * TWO_PI;
    float xv = xcum / (xlast + 1e-6f) * TWO_PI;
    float* pr = pos + (size_t)i * C_;
    const int nf = C_ / 2;  // 192
    for (int t = 0; t < nf / 2; ++t) {
        float dt = __powf(10000.f, (2.f * t) / nf);
        float sy, cy, sx_, cx;
        __sincosf(yv / dt, &sy, &cy);
        __sincosf(xv / dt, &sx_, &cx);
        pr[2 * t] = sy;
        pr[2 * t + 1] = cy;
        pr[nf + 2 * t] = sx_;
        pr[nf + 2 * t + 1] = cx;
    }
}

// ---------------------------------------------------------------------------
// K-means: init (top-K of score = first K in-group indices, then out-of-group)
// ---------------------------------------------------------------------------
__global__ void kmeans_init_kernel(const float* __restrict__ X, const float* __restrict__ w,
                                   float* __restrict__ centers, int K) {
    __shared__ int sel[64];
    int b = blockIdx.x;
    if (threadIdx.x == 0) {
        int cnt = 0;
        for (int n = 0; n < N_ && cnt < K; ++n)
            if (w[(size_t)b * N_ + n] > 0.5f) sel[cnt++] = n;
        for (int n = 0; n < N_ && cnt < K; ++n)
            if (w[(size_t)b * N_ + n] <= 0.5f) sel[cnt++] = n;
    }
    __syncthreads();
    for (int k = 0; k < K; ++k)
        for (int c = threadIdx.x; c < C_; c += blockDim.x)
            centers[((size_t)b * K + k) * C_ + c] = X[((size_t)b * N_ + sel[k]) * C_ + c];
}

__global__ void kmeans_assign_kernel(const float* __restrict__ X, const float* __restrict__ w,
                                     const float* __restrict__ centers, float* __restrict__ num,
                                     float* __restrict__ den, int K) {
    int i = blockIdx.x * blockDim.x + threadIdx.x;
    if (i >= BN_) return;
    int b = i / N_;
    const float* xr = X + (size_t)i * C_;
    float xsq = 0.f;
    for (int c = 0; c < C_; ++c) xsq += xr[c] * xr[c];
    const float* cb = centers + (size_t)b * K * C_;
    int best = 0;
    float bestd = 3.4e38f;
    for (int k = 0; k < K; ++k) {
        const float* cr = cb + (size_t)k * C_;
        float dot = 0.f, csq = 0.f;
        for (int c = 0; c < C_; ++c) {
            float cc = cr[c];
            dot += xr[c] * cc;
            csq += cc * cc;
        }
        float d = xsq - 2.f * dot + csq;
        if (d < bestd) { bestd = d; best = k; }
    }
    if (w[i] > 0.5f) {
        atomicAdd(&den[(size_t)b * K + best], 1.f);
        float* nr = num + ((size_t)b * K + best) * C_;
        for (int c = 0; c < C_; ++c) atomicAdd(&nr[c], xr[c]);
    }
}

__global__ void kmeans_update_kernel(const float* __restrict__ num, const float* __restrict__ den,
                                     float* __restrict__ centers, int K) {
    int i = blockIdx.x * blockDim.x + threadIdx.x;
    if (i >= B_ * K * C_) return;
    int bk = i / C_;
    float d = den[bk];
    if (d > 0.f) centers[i] = num[i] / fmaxf(d, 1e-6f);
}

// ---------------------------------------------------------------------------
// WMMA fp32 GEMM with TDM-staged weight tiles.
// D[M,Nc] = epilogue(A[M,K] (* rowmask) @ Wm[K,Nc] + bias + resid)
// Block = 4 waves = 64 rows x 16 cols. The shared 16-column weight slab
// (KB_ x 16, stride Nc, OOB columns zero-filled by the TDM) is DMA'd into LDS
// by wave 0 via tensor_load_to_lds + s_wait_tensorcnt, then all 4 waves feed
// V_WMMA_F32_16X16X4_F32 from LDS. EXEC stays all-1s (no divergent exits).
// ---------------------------------------------------------------------------
__global__ void gemm_wmma_f32(const float* __restrict__ A, const float* __restrict__ Wm,
                              const float* __restrict__ bias, const float* __restrict__ rowmask,
                              const float* __restrict__ resid, float* __restrict__ D,
                              int M, int K, int Nc, int fuse_relu) {
    __shared__ float sB[KB_ * 16];
    const int lane = threadIdx.x & 31;
    const int wid = threadIdx.x >> 5;
    const int mt = blockIdx.y * 4 + wid;   // 16-row tile (grid sized exactly: M % 64 == 0)
    const int nt = blockIdx.x;             // 16-col tile
    const int row0 = mt << 4;
    const int l15 = lane & 15;
    const int hi = lane >> 4;              // 0 or 1
    const int khalf = hi << 1;             // 0 or 2
    const int col = (nt << 4) + l15;
    const int arow = row0 + l15;
    const float am = rowmask ? rowmask[arow] : 1.f;
    const float* Ap = A + (size_t)arow * K + khalf;
    v8f acc = {};
    for (int k0 = 0; k0 < K; k0 += KB_) {
        __syncthreads();  // previous slab fully consumed before overwrite
#if __has_builtin(__builtin_amdgcn_tensor_load_to_lds)
        if (wid == 0) {
            // --- Tensor DMA descriptor (D#), ISA cdna5 ch.8 bit layout ---
            unsigned lds_off = (unsigned)(size_t)(void*)sB;  // generic LDS ptr: low 32b = offset
            unsigned long long ga =
                (unsigned long long)(size_t)(const void*)(Wm + (size_t)k0 * Nc + (nt << 4));
            int rc = Nc - (nt << 4);
            if (rc > 16) rc = 16;                      // edge tiles: TDM zero-fills cols >= rc
            unsigned td0 = (unsigned)rc;               // tensor_dim0 (elements)
            unsigned td1 = (unsigned)KB_;              // tensor_dim1
            unsigned long long st0 = (unsigned long long)(unsigned)Nc;  // dim0 stride (elements)
            u32x4 g0;
            g0[0] = 1u;                                               // count=1, user mode
            g0[1] = lds_off;                                          // lds_addr
            g0[2] = (unsigned)(ga & 0xFFFFFFFFull);                   // global_addr[31:0]
            g0[3] = (unsigned)((ga >> 32) & 0x01FFFFFFull) | 0x80000000u;  // addr[56:32]|type=2
            i32x8 g1;
            g1[0] = (int)(2u << 16);                                  // data_size = 4B
            g1[1] = (int)((td0 & 0xFFFFu) << 16);                     // tensor_dim0[15:0]
            g1[2] = (int)(((td0 >> 16) & 0xFFFFu) | ((td1 & 0xFFFFu) << 16));
            g1[3] = (int)(((td1 >> 16) & 0xFFFFu) | (16u << 16));     // tile_dim0 = 16
            g1[4] = (int)((unsigned)KB_ & 0xFFFFu);                   // tile_dim1 = KB_, dim2=0
            g1[5] = (int)(unsigned)(st0 & 0xFFFFFFFFull);             // dim0_stride[31:0]
            g1[6] = (int)(unsigned)((st0 >> 32) & 0xFFFFull);         // stride[47:32], d1s=0
            g1[7] = 0;
            i32x4 g2 = {0, 0, 0, 0};                                  // 2D tensor: unused
            i32x4 g3 = {0, 0, 0, 0};
            i32x8 g4 = {0, 0, 0, 0, 0, 0, 0, 0};                      // extra group (clang-23)
            __builtin_amdgcn_tensor_load_to_lds(g0, g1, g2, g3, g4, 0);
            __builtin_amdgcn_s_wait_tensorcnt(0);
        }
#else
        // Fallback: cooperative global->LDS staging with zero-fill on edge cols
        for (int i = threadIdx.x; i < KB_ * 16; i += 128) {
            int kr = i >> 4, cc = i & 15;
            int c2 = (nt << 4) + cc;
            sB[i] = (c2 < Nc) ? Wm[(size_t)(k0 + kr) * Nc + c2] : 0.f;
        }
#endif
        __syncthreads();  // slab visible to all 4 waves
#pragma unroll 4
        for (int kk = 0; kk < KB_; kk += 4) {
            v2f a = *(const v2f*)(Ap + k0 + kk);  // A[M=arow, K=k0+kk+khalf .. +1]
            a = a * am;
            v2f bfr;                               // B[K, N=col] from LDS (zero-filled edges)
            bfr.x = sB[(kk + khalf) * 16 + l15];
            bfr.y = sB[(kk + khalf + 1) * 16 + l15];
            acc = __builtin_amdgcn_wmma_f32_16x16x4_f32(false, a, false, bfr, (short)0, acc,
                                                        false, false);
        }
    }
    if (col < Nc) {
        const int rbase = row0 + (hi << 3);
#pragma unroll
        for (int r = 0; r < 8; ++r) {
            int row = rbase + r;
            float val = acc[r];
            if (bias) val += bias[col];
            if (resid) val += resid[(size_t)row * Nc + col];
            if (fuse_relu) val = fmaxf(val, 0.f);
            D[(size_t)row * Nc + col] = val;
        }
    }
}

// ---------------------------------------------------------------------------
// q = x + pos
// ---------------------------------------------------------------------------
__global__ void add_kernel(const float* __restrict__ a, const float* __restrict__ b,
                           float* __restrict__ o, int n) {
    int i = blockIdx.x * blockDim.x + threadIdx.x;
    if (i < n) o[i] = a[i] + b[i];
}

// softmax over P=9, in place; one thread per (b,n,head)
__global__ void softmax9_kernel(float* __restrict__ aw) {
    int i = blockIdx.x * blockDim.x + threadIdx.x;
    if (i >= BN_ * HEADS_) return;
    float* p = aw + (size_t)i * PTS_;
    float m = p[0];
#pragma unroll
    for (int j = 1; j < PTS_; ++j) m = fmaxf(m, p[j]);
    float e[PTS_], s = 0.f;
#pragma unroll
    for (int j = 0; j < PTS_; ++j) { e[j] = __expf(p[j] - m); s += e[j]; }
    float inv = 1.f / s;
#pragma unroll
    for (int j = 0; j < PTS_; ++j) p[j] = e[j] * inv;
}

// ---------------------------------------------------------------------------
// Deformable bilinear sample + per-point weighting. One wave32 per (b,n,head),
// lane = head-dim channel. attn[b,n,h*32+d] = sum_p aw * bilinear(v, loc(p))
// ---------------------------------------------------------------------------
__global__ void deform_sample_kernel(const float* __restrict__ V, const float* __restrict__ off,
                                     const float* __restrict__ aw, float* __restrict__ attn) {
    int item = blockIdx.x * (blockDim.x >> 5) + (threadIdx.x >> 5);
    if (item >= BN_ * HEADS_) return;
    int hIdx = item % HEADS_;
    int bn = item / HEADS_;
    int n = bn % N_;
    int b = bn / N_;
    int lane = threadIdx.x & 31;  // d channel
    int py = n / W_, px = n % W_;
    float refx = (px + 0.5f) / W_;
    float refy = (py + 0.5f) / H_;
    const float* vb = V + (size_t)b * N_ * C_ + hIdx * HD_ + lane;
    const float* offr = off + (size_t)bn * NOFF_ + hIdx * PTS_ * 2;
    const float* awr = aw + (size_t)bn * NAW_ + hIdx * PTS_;
    float out = 0.f;
#pragma unroll
    for (int p = 0; p < PTS_; ++p) {
        float lx = refx + offr[p * 2 + 0] / (float)W_;
        float ly = refy + offr[p * 2 + 1] / (float)H_;
        float Xc = lx * W_ - 0.5f;
        float Yc = ly * H_ - 0.5f;
        float x0f = floorf(Xc), y0f = floorf(Yc);
        float wx = Xc - x0f, wy = Yc - y0f;
        int x0 = (int)x0f, y0 = (int)y0f;
        float a = awr[p];
        float v00, v01, v10, v11;
        {
            int xi = x0, yi = y0;
            bool ok = (xi >= 0) & (xi < W_) & (yi >= 0) & (yi < H_);
            int xc = min(max(xi, 0), W_ - 1), yc = min(max(yi, 0), H_ - 1);
            v00 = ok ? vb[(size_t)(yc * W_ + xc) * C_] : 0.f;
        }
        {
            int xi = x0 + 1, yi = y0;
            bool ok = (xi >= 0) & (xi < W_) & (yi >= 0) & (yi < H_);
            int xc = min(max(xi, 0), W_ - 1), yc = min(max(yi, 0), H_ - 1);
            v01 = ok ? vb[(size_t)(yc * W_ + xc) * C_] : 0.f;
        }
        {
            int xi = x0, yi = y0 + 1;
            bool ok = (xi >= 0) & (xi < W_) & (yi >= 0) & (yi < H_);
            int xc = min(max(xi, 0), W_ - 1), yc = min(max(yi, 0), H_ - 1);
            v10 = ok ? vb[(size_t)(yc * W_ + xc) * C_] : 0.f;
        }
        {
            int xi = x0 + 1, yi = y0 + 1;
            bool ok = (xi >= 0) & (xi < W_) & (yi >= 0) & (yi < H_);
            int xc = min(max(xi, 0), W_ - 1), yc = min(max(yi, 0), H_ - 1);
            v11 = ok ? vb[(size_t)(yc * W_ + xc) * C_] : 0.f;
        }
        out += a * (v00 * (1.f - wx) * (1.f - wy) + v01 * wx * (1.f - wy) +
                    v10 * (1.f - wx) * wy + v11 * wx * wy);
    }
    attn[(size_t)bn * C_ + hIdx * HD_ + lane] = out;
}

// ---------------------------------------------------------------------------
// LayerNorm per row of C=384, eps=1e-5. One 128-thread block per row.
// ---------------------------------------------------------------------------
__global__ void layernorm_kernel(const float* __restrict__ Yb, const float* __restrict__ g,
                                 const float* __restrict__ bb, float* __restrict__ Xo) {
    __shared__ float red[128];
    const int row = blockIdx.x;
    const float* yr = Yb + (size_t)row * C_;
    float s = 0.f;
    for (int c = threadIdx.x; c < C_; c += 128) s += yr[c];
    red[threadIdx.x] = s;
    __syncthreads();
    for (int o = 64; o > 0; o >>= 1) {
        if (threadIdx.x < o) red[threadIdx.x] += red[threadIdx.x + o];
        __syncthreads();
    }
    const float mean = red[0] * (1.f / C_);
    __syncthreads();
    float vs = 0.f;
    for (int c = threadIdx.x; c < C_; c += 128) {
        float d = yr[c] - mean;
        vs += d * d;
    }
    red[threadIdx.x] = vs;
    __syncthreads();
    for (int o = 64; o > 0; o >>= 1) {
        if (threadIdx.x < o) red[threadIdx.x] += red[threadIdx.x + o];
        __syncthreads();
    }
    const float rstd = rsqrtf(red[0] * (1.f / C_) + 1e-5f);
    for (int c = threadIdx.x; c < C_; c += 128)
        Xo[(size_t)row * C_ + c] = (yr[c] - mean) * rstd * g[c] + bb[c];
}

// ---------------------------------------------------------------------------
// out[b, 0:N)=x, [N:N+20)=fg centers, [N+20:N+70)=bg centers
// ---------------------------------------------------------------------------
__global__ void concat_kernel(const float* __restrict__ X, const float* __restrict__ fgc,
                              const float* __restrict__ bgc, float* __restrict__ out,
                              int total) {
    int i = blockIdx.x * blockDim.x + threadIdx.x;
    if (i >= total) return;
    int c = i % C_;
    int r = i / C_;
    int b = r / NOUT_ROWS_;
    int rr = r % NOUT_ROWS_;
    float v;
    if (rr < N_)
        v = X[((size_t)b * N_ + rr) * C_ + c];
    else if (rr < N_ + KFG_)
        v = fgc[((size_t)b * KFG_ + (rr - N_)) * C_ + c];
    else
        v = bgc[((size_t)b * KBG_ + (rr - N_ - KFG_)) * C_ + c];
    out[i] = v;
}

// ---------------------------------------------------------------------------
extern "C" void kernel_launch(void* const* d_in, const int* in_sizes, int n_in,
                              void* d_out, int out_size, void* d_ws, size_t ws_size,
                              hipStream_t stream) {
    const float* s_x   = (const float*)d_in[0];
    const float* s_pad = (const float*)d_in[1];
    const float* s_sup = (const float*)d_in[2];
    const float* Wv    = (const float*)d_in[3];
    const float* bv    = (const float*)d_in[4];
    const float* Woff  = (const float*)d_in[5];
    const float* boff  = (const float*)d_in[6];
    const float* Wa    = (const float*)d_in[7];
    const float* ba    = (const float*)d_in[8];
    const float* Wout  = (const float*)d_in[9];
    const float* bout  = (const float*)d_in[10];
    const float* ln1g  = (const float*)d_in[11];
    const float* ln1b  = (const float*)d_in[12];
    const float* Wf1   = (const float*)d_in[13];
    const float* Wf2   = (const float*)d_in[14];
    const float* ln2g  = (const float*)d_in[15];
    const float* ln2b  = (const float*)d_in[16];

    const size_t BNC = (size_t)BN_ * C_;
    float* ws = (float*)d_ws;
    size_t o = 0;
    auto alloc = [&](size_t n) { float* p = ws + o; o += n; return p; };
    float* X    = alloc(BNC);
    float* Q    = alloc(BNC);                 // reused as GEMM residual target Y
    float* POS  = alloc(BNC);
    float* V    = alloc(BNC);
    float* HBUF = alloc((size_t)BN_ * FFN_);  // ATTN aliases first BNC of this
    float* OFF  = alloc((size_t)BN_ * NOFF_);
    float* AW   = alloc((size_t)BN_ * NAW_);
    float* NOTP = alloc(BN_);
    float* WFGm = alloc(BN_);
    float* WBGm = alloc(BN_);
    float* FGC  = alloc((size_t)B_ * KFG_ * C_);
    float* BGC  = alloc((size_t)B_ * KBG_ * C_);
    float* KNUM = alloc((size_t)B_ * KBG_ * C_);
    float* KDEN = alloc((size_t)B_ * KBG_);
    float* Y    = Q;        // Q dead once OFF/AW computed
    float* ATTN = HBUF;     // attn dead before FFN hidden is written

    (void)in_sizes; (void)n_in; (void)ws_size;

    // --- prep ---
    transpose_kernel<<<(int)((BNC + 255) / 256), 256, 0, stream>>>(s_x, X);
    mask_kernel<<<(BN_ + 255) / 256, 256, 0, stream>>>(s_pad, s_sup, NOTP, WFGm, WBGm);
    pos_kernel<<<(BN_ + 255) / 256, 256, 0, stream>>>(NOTP, POS);

    // --- masked k-means (fg K=20, bg K=50), 10 Lloyd iterations each ---
    struct { float* w; float* ctr; int K; } grp[2] = {{WFGm, FGC, KFG_}, {WBGm, BGC, KBG_}};
    for (int g = 0; g < 2; ++g) {
        kmeans_init_kernel<<<B_, 64, 0, stream>>>(X, grp[g].w, grp[g].ctr, grp[g].K);
        for (int it = 0; it < NITER_; ++it) {
            (void)hipMemsetAsync(KNUM, 0, (size_t)B_ * grp[g].K * C_ * sizeof(float), stream);
            (void)hipMemsetAsync(KDEN, 0, (size_t)B_ * grp[g].K * sizeof(float), stream);
            kmeans_assign_kernel<<<(BN_ + 127) / 128, 128, 0, stream>>>(X, grp[g].w, grp[g].ctr,
                                                                        KNUM, KDEN, grp[g].K);
            int nUpd = B_ * grp[g].K * C_;
            kmeans_update_kernel<<<(nUpd + 255) / 256, 256, 0, stream>>>(KNUM, KDEN, grp[g].ctr,
                                                                         grp[g].K);
        }
    }

    // --- 2 deformable-attention layers ---
    auto gemm = [&](const float* A, const float* Wm, const float* bias, const float* rowmask,
                    const float* resid, float* D, int M, int K, int Nc, int relu) {
        dim3 grid((Nc + 15) / 16, M / 64);
        gemm_wmma_f32<<<grid, 128, 0, stream>>>(A, Wm, bias, rowmask, resid, D, M, K, Nc, relu);
    };

    for (int l = 0; l < LAYERS_; ++l) {
        // q = x + pos
        add_kernel<<<(int)((BNC + 255) / 256), 256, 0, stream>>>(X, POS, Q, (int)BNC);
        // v = (x * notpad) @ Wv + bv
        gemm(X, Wv + (size_t)l * C_ * C_, bv + (size_t)l * C_, NOTP, nullptr, V, BN_, C_, C_, 0);
        // off = q @ Woff + boff ; aw_logits = q @ Wa + ba
        gemm(Q, Woff + (size_t)l * C_ * NOFF_, boff + (size_t)l * NOFF_, nullptr, nullptr, OFF,
             BN_, C_, NOFF_, 0);
        gemm(Q, Wa + (size_t)l * C_ * NAW_, ba + (size_t)l * NAW_, nullptr, nullptr, AW,
             BN_, C_, NAW_, 0);
        softmax9_kernel<<<(BN_ * HEADS_ + 255) / 256, 256, 0, stream>>>(AW);
        // bilinear deformable sampling + per-point weighting
        {
            int items = BN_ * HEADS_;
            deform_sample_kernel<<<(items + 3) / 4, 128, 0, stream>>>(V, OFF, AW, ATTN);
        }
        // y = x + attn @ Wout + bout ; x = LN1(y)
        gemm(ATTN, Wout + (size_t)l * C_ * C_, bout + (size_t)l * C_, nullptr, X, Y,
             BN_, C_, C_, 0);
        layernorm_kernel<<<BN_, 128, 0, stream>>>(Y, ln1g + (size_t)l * C_, ln1b + (size_t)l * C_,
                                                  X);
        // h = relu(x @ Wf1) ; y = x + h @ Wf2 ; x = LN2(y)
        gemm(X, Wf1 + (size_t)l * C_ * FFN_, nullptr, nullptr, nullptr, HBUF, BN_, C_, FFN_, 1);
        gemm(HBUF, Wf2 + (size_t)l * FFN_ * C_, nullptr, nullptr, X, Y, BN_, FFN_, C_, 0);
        layernorm_kernel<<<BN_, 128, 0, stream>>>(Y, ln2g + (size_t)l * C_, ln2b + (size_t)l * C_,
                                                  X);
    }

    // --- concat [x, fg_c, bg_c] ---
    concat_kernel<<<(out_size + 255) / 256, 256, 0, stream>>>(X, FGC, BGC, (float*)d_out,
                                                              out_size);
}